// Four_Pos_Fusion_Embedding_16320875725026
// MI455X (gfx1250) — compile-verified
//
#include <hip/hip_runtime.h>

typedef __attribute__((ext_vector_type(2))) float v2f;
typedef __attribute__((ext_vector_type(4))) float v4f;
typedef __attribute__((ext_vector_type(8))) float v8f;

namespace {
constexpr int kB = 4;
constexpr int kL = 200;
constexpr int kH = 160;       // hidden size
constexpr int kT = 1024;      // table rows (2*MAX_SEQ_LEN)
constexpr int kShift = 512;   // MAX_SEQ_LEN
constexpr int kH4 = kH / 4;   // 40 float4 per row
}

// ---------------------------------------------------------------------------
// Kernel 1: P[z] = pe[z] (1024x160) @ W[z*160:(z+1)*160, :] (160x160)
// One wave per 16x16 output tile; V_WMMA_F32_16X16X4_F32, K-loop of 40.
// grid = (64 M-tiles, 10 N-tiles, 4 tables), block = 32 (one wave, EXEC all 1s)
// ---------------------------------------------------------------------------
__global__ __launch_bounds__(32)
void table_gemm_wmma(const float* __restrict__ pe0, const float* __restrict__ pe1,
                     const float* __restrict__ pe2, const float* __restrict__ pe3,
                     const float* __restrict__ W, float* __restrict__ P)
{
    const int z     = blockIdx.z;
    const float* pe = (z == 0) ? pe0 : (z == 1) ? pe1 : (z == 2) ? pe2 : pe3;

    const int mBase = blockIdx.x * 16;          // 0..1008
    const int nBase = blockIdx.y * 16;          // 0..144
    const int lane  = threadIdx.x;              // 0..31
    const int lo    = lane & 15;
    const int hi    = lane >> 4;                // 0 or 1

    // A: row (mBase+lo) of pe, lane's K pair starts at k + 2*hi
    const float* arow = pe + (size_t)(mBase + lo) * kH;
    // B: W block for table z, column (nBase+lo)
    const float* bcol = W + (size_t)(z * kH) * kH + nBase + lo;

    v8f acc = {};
    #pragma unroll 4
    for (int k = 0; k < kH; k += 4) {
        const int kk = k + 2 * hi;
        v2f a, b;
        a.x = arow[kk + 0];
        a.y = arow[kk + 1];
        b.x = bcol[(size_t)(kk + 0) * kH];
        b.y = bcol[(size_t)(kk + 1) * kH];
        // D = A*B + C   (8 args: neg_a, A, neg_b, B, c_mod, C, reuse_a, reuse_b)
        acc = __builtin_amdgcn_wmma_f32_16x16x4_f32(false, a, false, b,
                                                    (short)0, acc, false, false);
    }

    // C/D layout: VGPR v holds row M = v + 8*hi, col N = lo
    float* ptile = P + (size_t)z * kT * kH + (size_t)mBase * kH + nBase;
    #pragma unroll
    for (int v = 0; v < 8; ++v) {
        ptile[(size_t)(v + 8 * hi) * kH + lo] = acc[v];
    }
}

// ---------------------------------------------------------------------------
// Kernel 2: out[b,i,j,:] = relu(Pss[iss] + Pse[ise] + Pes[ies] + Pee[iee] + bias)
// One thread per float4 of output; writes fully coalesced, gathers hit L2
// (P tables total 2.6 MB << 192 MB L2).
// ---------------------------------------------------------------------------
__global__ __launch_bounds__(256)
void gather_fuse_relu(const float* __restrict__ P,
                      const int* __restrict__ pos_s, const int* __restrict__ pos_e,
                      const float* __restrict__ bias, float* __restrict__ out)
{
    const unsigned total = (unsigned)kB * kL * kL * kH4;   // 6,400,000
    unsigned tid = blockIdx.x * blockDim.x + threadIdx.x;
    if (tid >= total) return;

    unsigned h4 = tid % kH4;
    unsigned t  = tid / kH4;
    unsigned j  = t % kL;  t /= kL;
    unsigned i  = t % kL;  t /= kL;
    unsigned b  = t;

    const int ps_i = pos_s[b * kL + i];
    const int ps_j = pos_s[b * kL + j];
    const int pe_i = pos_e[b * kL + i];
    const int pe_j = pos_e[b * kL + j];

    const int iss = ps_i - ps_j + kShift;
    const int ise = ps_i - pe_j + kShift;
    const int ies = pe_i - ps_j + kShift;
    const int iee = pe_i - pe_j + kShift;

    const v4f* Pss = (const v4f*)(P + 0 * (size_t)kT * kH);
    const v4f* Pse = (const v4f*)(P + 1 * (size_t)kT * kH);
    const v4f* Pes = (const v4f*)(P + 2 * (size_t)kT * kH);
    const v4f* Pee = (const v4f*)(P + 3 * (size_t)kT * kH);
    const v4f* bv  = (const v4f*)bias;

    v4f s = Pss[(unsigned)iss * kH4 + h4]
          + Pse[(unsigned)ise * kH4 + h4]
          + Pes[(unsigned)ies * kH4 + h4]
          + Pee[(unsigned)iee * kH4 + h4]
          + bv[h4];

    v4f r;
    r.x = fmaxf(s.x, 0.0f);
    r.y = fmaxf(s.y, 0.0f);
    r.z = fmaxf(s.z, 0.0f);
    r.w = fmaxf(s.w, 0.0f);

    ((v4f*)out)[tid] = r;
}

// ---------------------------------------------------------------------------
// Host launcher
// Inputs (setup_inputs order): pe_ss, pe_se, pe_es, pe_ee, W, b, pos_s, pos_e
// d_out: [4,200,200,160] float32.  d_ws: needs 4*1024*160*4 = 2,621,440 bytes.
// ---------------------------------------------------------------------------
extern "C" void kernel_launch(void* const* d_in, const int* in_sizes, int n_in,
                              void* d_out, int out_size, void* d_ws, size_t ws_size,
                              hipStream_t stream)
{
    const float* pe_ss = (const float*)d_in[0];
    const float* pe_se = (const float*)d_in[1];
    const float* pe_es = (const float*)d_in[2];
    const float* pe_ee = (const float*)d_in[3];
    const float* W     = (const float*)d_in[4];
    const float* bias  = (const float*)d_in[5];
    const int*   pos_s = (const int*)d_in[6];
    const int*   pos_e = (const int*)d_in[7];

    float* P = (float*)d_ws;   // [4][1024][160] premultiplied tables

    // Kernel 1: 4 table GEMMs via f32 WMMA
    dim3 g1(kT / 16, kH / 16, 4);   // (64, 10, 4)
    table_gemm_wmma<<<g1, 32, 0, stream>>>(pe_ss, pe_se, pe_es, pe_ee, W, P);

    // Kernel 2: gather + sum + bias + ReLU, one thread per float4
    const unsigned total = (unsigned)kB * kL * kL * kH4;
    gather_fuse_relu<<<(total + 255) / 256, 256, 0, stream>>>(P, pos_s, pos_e, bias,
                                                              (float*)d_out);
}